// KnowledgeGraphConvLayer_nBatch_geometric_78950088835525
// MI455X (gfx1250) — compile-verified
//
#include <hip/hip_runtime.h>
#include <hip/hip_bf16.h>

// CDNA5 (gfx1250) WMMA types
typedef __attribute__((ext_vector_type(2))) float v2f;
typedef __attribute__((ext_vector_type(8))) float v8f;

__device__ __forceinline__ float lrelu(float x) { return x > 0.0f ? x : 0.01f * x; }

// ---------------------------------------------------------------------------
// GEMM: out[N x 64] = op_in(A[N x 64]) @ W[64 x 64]  (+ bias) (op_out)
// One wave computes a 16-row strip using v_wmma_f32_16x16x4_f32 (K=4 x 16 steps).
// W is pre-swizzled into LDS in per-lane B-fragment order so each (nt,kk) step
// is a single conflict-free ds_load_b64 directly into an even VGPR pair:
//   bfrag[(nt*16+kk)*32 + lane] = { W[4kk+koff][col], W[4kk+koff+1][col] }
//   with koff = 2*(lane>=16), col = nt*16 + (lane&15)      (ISA B 4x16 layout)
// A 16x4 frag = v2f/lane (lane&15 = M row; lane>=16 -> K+2)  (ISA A 16x4 layout)
// C/D 16x16  = v8f/lane (vgpr r: M = r + 8*(lane>=16), N = lane&15 in tile)
// ---------------------------------------------------------------------------
template <bool LEAKY_IN, bool LEAKY_OUT, bool HAS_BIAS>
__global__ __launch_bounds__(256) void gemm64_wmma_kernel(
    const float* __restrict__ A, const float* __restrict__ W,
    const float* __restrict__ bias, float* __restrict__ out, int N) {
  __shared__ v2f bfrag[2048];  // 16 KB: 4 col-tiles x 16 k-steps x 32 lanes

  const int t = threadIdx.x;
  // Stage W into swizzled B-fragment layout (one-time 16 KB shuffle).
#pragma unroll
  for (int j = 0; j < 8; ++j) {
    const int o = t + j * 256;
    const int lo = o & 31;            // target lane
    const int kk = (o >> 5) & 15;     // k-step
    const int nt = o >> 9;            // col tile
    const int row = kk * 4 + ((lo >> 4) << 1);
    const int col = nt * 16 + (lo & 15);
    v2f b;
    b.x = W[row * 64 + col];
    b.y = W[(row + 1) * 64 + col];
    bfrag[o] = b;
  }
  __syncthreads();

  const int wave = t >> 5;
  const int lane = t & 31;
  const int row_base = (blockIdx.x * 8 + wave) * 16;
  if (row_base >= N) return;  // whole-wave uniform: EXEC stays all-ones

  const int l15 = lane & 15;
  const int hi = lane >> 4;  // 0 for lanes 0-15, 1 for lanes 16-31

  if (row_base + 16 <= N) {
    // ---- full 16-row strip: WMMA path ----
    v2f afrag[16];
    {
      const float* arow = A + (size_t)(row_base + l15) * 64;
      const v2f* arow2 = (const v2f*)arow;  // 8B-aligned (row base 256B)
#pragma unroll
      for (int kk = 0; kk < 16; ++kk) {
        v2f a = arow2[kk * 2 + hi];
        if (LEAKY_IN) { a.x = lrelu(a.x); a.y = lrelu(a.y); }
        afrag[kk] = a;
      }
    }
    const int mbase = row_base + hi * 8;
#pragma unroll
    for (int nt = 0; nt < 4; ++nt) {
      const int col = nt * 16 + l15;
      const float cinit = HAS_BIAS ? bias[col] : 0.0f;
      v8f acc;
#pragma unroll
      for (int r = 0; r < 8; ++r) acc[r] = cinit;
      const v2f* bl = &bfrag[nt * 16 * 32 + lane];
#pragma unroll
      for (int kk = 0; kk < 16; ++kk) {
        const v2f b = bl[kk * 32];  // single ds_load_b64, bank-conflict-free
        acc = __builtin_amdgcn_wmma_f32_16x16x4_f32(
            /*neg_a=*/false, afrag[kk], /*neg_b=*/false, b,
            /*c_mod=*/(short)0, acc, /*reuse_a=*/false, /*reuse_b=*/false);
      }
#pragma unroll
      for (int r = 0; r < 8; ++r) {
        float v = acc[r];
        if (LEAKY_OUT) v = lrelu(v);
        out[(size_t)(mbase + r) * 64 + col] = v;
      }
    }
  } else {
    // ---- tail strip (< 16 rows): scalar fallback, W read from global ----
    for (int r = row_base + lane; r < N; r += 32) {
      const float* ar = A + (size_t)r * 64;
      for (int c = 0; c < 64; ++c) {
        float s = HAS_BIAS ? bias[c] : 0.0f;
#pragma unroll
        for (int k = 0; k < 64; ++k) {
          float a = ar[k];
          if (LEAKY_IN) a = lrelu(a);
          s += a * W[k * 64 + c];
        }
        if (LEAKY_OUT) s = lrelu(s);
        out[(size_t)r * 64 + c] = s;
      }
    }
  }
}

// agg[i][f] = b[f]  (bias-initialized accumulator so scatter can pure-add)
__global__ void init_bias_kernel(float* __restrict__ agg,
                                 const float* __restrict__ b, int total) {
  int idx = blockIdx.x * blockDim.x + threadIdx.x;
  if (idx < total) agg[idx] = b[idx & 63];
}

// One thread per (edge, feature): agg[dst[e]][f] += h[src[e]][f] * ew[e]
// h (25.6 MB) is L2-resident on MI455X (192 MB L2) -> gathers/atomics hit L2.
__global__ void scatter_kernel(const float* __restrict__ h,
                               const long long* __restrict__ src,
                               const long long* __restrict__ dst,
                               const float* __restrict__ ew,
                               float* __restrict__ agg, int E) {
  long long tid = (long long)blockIdx.x * blockDim.x + threadIdx.x;
  if (tid >= (long long)E * 64) return;
  const int e = (int)(tid >> 6);
  const int f = (int)(tid & 63);
  const long long s = src[e];
  const long long d = dst[e];
  const float v = h[s * 64 + f] * ew[e];
  atomicAdd(&agg[d * 64 + f], v);
}

extern "C" void kernel_launch(void* const* d_in, const int* in_sizes, int n_in,
                              void* d_out, int out_size, void* d_ws, size_t ws_size,
                              hipStream_t stream) {
  (void)n_in; (void)out_size; (void)ws_size;
  // setup_inputs order:
  // 0 features[N,64] f32 | 1 adjs[2,E] i64 | 2 values[E] f32 | 3 N | 4 E
  // 5 W1[64,64] | 6 b1[64] | 7 W2[64,64] | 8 b2[64] | 9 Wl[64,64] | 10 bl[64]
  const float* X  = (const float*)d_in[0];
  const long long* adjs = (const long long*)d_in[1];  // reference dtype is int64
  const float* ew = (const float*)d_in[2];
  const float* W1 = (const float*)d_in[5];
  const float* b1 = (const float*)d_in[6];
  const float* W2 = (const float*)d_in[7];
  const float* b2 = (const float*)d_in[8];
  const float* Wl = (const float*)d_in[9];
  const float* bl = (const float*)d_in[10];
  float* out = (float*)d_out;

  const int N = in_sizes[0] / 64;
  const int E = in_sizes[2];
  const long long* src = adjs;
  const long long* dst = adjs + E;

  float* h   = (float*)d_ws;             // N*64 f32
  float* agg = h + (size_t)N * 64;       // N*64 f32

  const int gemm_grid = (N + 127) / 128;            // 8 waves x 16 rows per block
  const int init_grid = (N * 64 + 255) / 256;
  const int scat_grid = (int)(((long long)E * 64 + 255) / 256);

  // Layer 1: h = X@W1 ; agg = b1 + scatter(ew * h[src]) ; (leaky fused into next A-load)
  gemm64_wmma_kernel<false, false, false><<<gemm_grid, 256, 0, stream>>>(X, W1, nullptr, h, N);
  init_bias_kernel<<<init_grid, 256, 0, stream>>>(agg, b1, N * 64);
  scatter_kernel<<<scat_grid, 256, 0, stream>>>(h, src, dst, ew, agg, E);

  // Layer 2: h = leaky(agg)@W2 ; agg = b2 + scatter(ew * h[src])
  gemm64_wmma_kernel<true, false, false><<<gemm_grid, 256, 0, stream>>>(agg, W2, nullptr, h, N);
  init_bias_kernel<<<init_grid, 256, 0, stream>>>(agg, b2, N * 64);
  scatter_kernel<<<scat_grid, 256, 0, stream>>>(h, src, dst, ew, agg, E);

  // Head: out = leaky(agg @ Wl + bl)
  gemm64_wmma_kernel<false, true, true><<<gemm_grid, 256, 0, stream>>>(agg, Wl, bl, out, N);
}